// LabelSmoothing_62792421868006
// MI455X (gfx1250) — compile-verified
//
#include <hip/hip_runtime.h>
#include <math.h>

typedef float v2f __attribute__((ext_vector_type(2)));
typedef float v4f __attribute__((ext_vector_type(4)));
typedef float v8f __attribute__((ext_vector_type(8)));

// One wave32 per row: streaming NT float4 loads (4-deep unrolled for MLP),
// per-lane fp32 accumulation, cross-lane reduction via V_WMMA_F32_16X16X4_F32
// with B = ones.
__global__ __launch_bounds__(256)
void ls_row_kernel(const float* __restrict__ x, const int* __restrict__ target,
                   float* __restrict__ partial, int N, int V,
                   float s, float crow) {
    const int lane = threadIdx.x & 31;
    const int row  = blockIdx.x * 8 + (threadIdx.x >> 5);
    if (row >= N) return;  // wave-uniform: EXEC stays all-ones for live waves

    const float* __restrict__ xrow = x + (size_t)row * (size_t)V;
    const v4f* __restrict__ xrow4 = (const v4f*)xrow;
    const int V4 = V >> 2;

    // Main streaming loop, unrolled x4: 4 outstanding b128 NT loads per wave,
    // 2 KB contiguous per iteration, 4 independent accumulators.
    v4f acc0 = {0.f,0.f,0.f,0.f}, acc1 = {0.f,0.f,0.f,0.f};
    v4f acc2 = {0.f,0.f,0.f,0.f}, acc3 = {0.f,0.f,0.f,0.f};
    int i = lane;
    for (; i + 96 < V4; i += 128) {
        v4f a0 = __builtin_nontemporal_load(xrow4 + i);
        v4f a1 = __builtin_nontemporal_load(xrow4 + i + 32);
        v4f a2 = __builtin_nontemporal_load(xrow4 + i + 64);
        v4f a3 = __builtin_nontemporal_load(xrow4 + i + 96);
        acc0 += a0; acc1 += a1; acc2 += a2; acc3 += a3;
    }
    for (; i < V4; i += 32)
        acc0 += __builtin_nontemporal_load(xrow4 + i);

    // Scalar tail (V % 4 != 0 robustness; no-op for V = 32000).
    float accs = 0.f;
    for (int j = (V4 << 2) + lane; j < V; j += 32) accs += xrow[j];

    v4f acc4 = (acc0 + acc1) + (acc2 + acc3);
    float p = (acc4.x + acc4.y) + (acc4.z + acc4.w) + accs;

    // Cross-lane sum via WMMA: A[m,0]=p_m (lanes 0-15), A[m,2]=p_{m+16}
    // (lanes 16-31), B = all ones  =>  D[m,n] = p_m + p_{m+16}.
    v2f a; a.x = p;   a.y = 0.0f;
    v2f b; b.x = 1.0f; b.y = 1.0f;
    v8f c = {};
    v8f d = __builtin_amdgcn_wmma_f32_16x16x4_f32(
        /*neg_a=*/false, a, /*neg_b=*/false, b,
        /*c_mod=*/(short)0, c, /*reuse_a=*/false, /*reuse_b=*/false);

    // lane 0 holds D[m=0..7, 0] in d[0..7]; lane 16 holds D[m=8..15, 0].
    float t = ((d[0] + d[1]) + (d[2] + d[3])) + ((d[4] + d[5]) + (d[6] + d[7]));
    float t_hi = __shfl_down(t, 16, 32);

    if (lane == 0) {
        float rowsum = t + t_hi;
        int   tg = target[row];
        float x0 = xrow[0];
        float xt = xrow[tg];
        float contrib = (tg != 0)
            ? (crow - s * rowsum + s * x0 + (s - 0.9f) * xt)
            : 0.0f;
        partial[row] = contrib;
    }
}

// Deterministic fixed-order final reduction of N row partials.
__global__ __launch_bounds__(256)
void ls_reduce_kernel(const float* __restrict__ partial, float* __restrict__ out, int N) {
    __shared__ float sm[256];
    float a = 0.f;
    for (int i = threadIdx.x; i < N; i += 256) a += partial[i];
    sm[threadIdx.x] = a;
    __syncthreads();
    for (int off = 128; off > 0; off >>= 1) {
        if ((int)threadIdx.x < off) sm[threadIdx.x] += sm[threadIdx.x + off];
        __syncthreads();
    }
    if (threadIdx.x == 0) out[0] = sm[0];
}

extern "C" void kernel_launch(void* const* d_in, const int* in_sizes, int n_in,
                              void* d_out, int out_size, void* d_ws, size_t ws_size,
                              hipStream_t stream) {
    const float* x      = (const float*)d_in[0];
    const int*   target = (const int*)d_in[1];
    float*       out    = (float*)d_out;
    float*       partial = (float*)d_ws;  // N floats of scratch

    const int N = in_sizes[1];
    const int V = (int)((long long)in_sizes[0] / (long long)N);

    const float s    = 0.1f / (float)(V - 2);
    // (V-2)*s*log(s) + 0.9*log(0.9) == 0.1*log(s) + 0.9*log(0.9)
    const float crow = 0.1f * logf(s) + 0.9f * logf(0.9f);

    const int waves_per_block = 8;  // 256 threads = 8 wave32
    const int blocks = (N + waves_per_block - 1) / waves_per_block;

    hipLaunchKernelGGL(ls_row_kernel, dim3(blocks), dim3(256), 0, stream,
                       x, target, partial, N, V, s, crow);
    hipLaunchKernelGGL(ls_reduce_kernel, dim3(1), dim3(256), 0, stream,
                       partial, out, N);
}